// FixedLiftPDEGCNN_87557203296305
// MI455X (gfx1250) — compile-verified
//
#include <hip/hip_runtime.h>

// ---------------------------------------------------------------------------
// PDE-G-CNN forward (cakewavelet lift + 4 CDE layers) for MI455X / gfx1250.
// FFT-as-GEMM + channel mixers use v_wmma_f32_16x16x4_f32 (wave32 WMMA).
// Morphology / convection are bilinear-gather VALU kernels; everything (~90MB)
// is L2-resident on the 192MB L2, so we stay in f32 end-to-end.
// ---------------------------------------------------------------------------

#define NF    192
#define HW2   36864            // NF*NF
#define ORN   8
#define NB    2
#define C1N   40
#define C2N   16
#define VOLN  294912           // ORN*HW2
#define PTSN  589824           // NB*VOLN
#define CDE_ELEMS 9437184LL    // NB*C2N*VOLN

#define DTH      0.7853981633974483f
#define PI_F     3.14159265358979323846f
#define TWO_PI_F 6.283185307179586f

// ---- workspace layout (float offsets) ----
static const long long OFF_C    = 0;
static const long long OFF_S    = OFF_C    + HW2;
static const long long OFF_PSI  = OFF_S    + HW2;
static const long long OFF_U0   = OFF_PSI  + (long long)ORN*HW2;
static const long long OFF_TRE  = OFF_U0   + PTSN;
static const long long OFF_TIM  = OFF_TRE  + (long long)NB*HW2;
static const long long OFF_XHRE = OFF_TIM  + (long long)NB*HW2;
static const long long OFF_XHIM = OFF_XHRE + (long long)NB*HW2;
static const long long OFF_YRE  = OFF_XHIM + (long long)NB*HW2;
static const long long OFF_YIM  = OFF_YRE  + 16LL*HW2;
static const long long OFF_P    = OFF_YIM  + 16LL*HW2;
static const long long OFF_Q    = OFF_P    + 16LL*HW2;
static const long long OFF_STAT = OFF_Q    + 16LL*HW2;   // 512 floats = 256 doubles
static const long long OFF_M2   = OFF_STAT + 512;
static const long long OFF_B2   = OFF_M2   + C2N*C1N;
static const long long OFF_KTAB = OFF_B2   + C2N;
static const long long OFF_BUFA = OFF_KTAB + C2N*125 + 8;
static const long long OFF_BUFB = OFF_BUFA + CDE_ELEMS;
static const long long WS_FLOATS= OFF_BUFB + CDE_ELEMS;

typedef float v2f __attribute__((ext_vector_type(2)));
typedef float v8f __attribute__((ext_vector_type(8)));

// ---------------------------------------------------------------------------
// Generic strided-batched GEMM:  D = scale*(A1@B1 + s2*A2@B2) [+ bias]
// via V_WMMA_F32_16X16X4_F32. One 16x16 D tile per wave.
// Fragment layout (ISA 7.12.2): A: lane m=l&15 holds K=2h,2h+1 (h=l>>4);
// B: lane n=l&15 holds col n, K=2h,2h+1; C/D: row = r + 8h, col = l&15.
// f32 WMMA has no A/B NEG modifier (probe: valid range [0,0]) -> fold the
// sign s2 (+-1.0) into the B2 fragment with VALU muls.
// ---------------------------------------------------------------------------
struct GemmArgs {
  const float* A1; const float* A2; const float* B1; const float* B2;
  const float* bias; float* D;
  int K, lda, ldb, ldd;
  long long sA, sB, sD;
  float s2, scale;
};

__global__ __launch_bounds__(128) void gemm2_wmma(GemmArgs g, int tilesM, int tilesN, int batch) {
  int wave = (int)((blockIdx.x * blockDim.x + threadIdx.x) >> 5);
  int lane = threadIdx.x & 31;
  int tilesPer = tilesM * tilesN;
  if (wave >= tilesPer * batch) return;             // wave-uniform: EXEC stays all-1
  int bb = wave / tilesPer;
  int t  = wave - bb * tilesPer;
  int tm = t / tilesN, tn = t - tm * tilesN;
  int h  = lane >> 4;                               // K-pair selector
  int mn = lane & 15;
  const float* A1 = g.A1 + (long long)bb * g.sA;
  const float* B1 = g.B1 + (long long)bb * g.sB;
  const float* A2 = g.A2 ? g.A2 + (long long)bb * g.sA : nullptr;
  const float* B2 = g.B2 ? g.B2 + (long long)bb * g.sB : nullptr;
  int rowA = tm * 16 + mn;
  int colB = tn * 16 + mn;
  v8f acc = {0.f,0.f,0.f,0.f,0.f,0.f,0.f,0.f};
  float sgn2 = g.s2;                                // +-1.0, wave-uniform
  for (int kb = 0; kb < g.K; kb += 4) {
    int k0 = kb + 2 * h;
    v2f a1, b1f;
    a1.x  = A1[(long long)rowA * g.lda + k0];
    a1.y  = A1[(long long)rowA * g.lda + k0 + 1];
    b1f.x = B1[(long long)k0 * g.ldb + colB];
    b1f.y = B1[(long long)(k0 + 1) * g.ldb + colB];
    acc = __builtin_amdgcn_wmma_f32_16x16x4_f32(false, a1, false, b1f, (short)0, acc, false, false);
    if (A2) {
      v2f a2, b2f;
      a2.x  = A2[(long long)rowA * g.lda + k0];
      a2.y  = A2[(long long)rowA * g.lda + k0 + 1];
      b2f.x = sgn2 * B2[(long long)k0 * g.ldb + colB];
      b2f.y = sgn2 * B2[(long long)(k0 + 1) * g.ldb + colB];
      acc = __builtin_amdgcn_wmma_f32_16x16x4_f32(false, a2, false, b2f, (short)0, acc, false, false);
    }
  }
  float* D = g.D + (long long)bb * g.sD;
  #pragma unroll
  for (int r = 0; r < 8; ++r) {
    int m = tm * 16 + r + 8 * h;
    float v = g.scale * acc[r];
    if (g.bias) v += g.bias[m];
    D[(long long)m * g.ldd + colB] = v;
  }
}

// ---------------------------------------------------------------------------
// Interpolation helpers (map_coordinates order=1, mode='nearest' semantics)
// ---------------------------------------------------------------------------
__device__ __forceinline__ float bilin(const float* __restrict__ p, float cy, float cx) {
  int y0 = (int)cy, x0 = (int)cx;                   // coords already in [0, NF-1]
  float fy = cy - (float)y0, fx = cx - (float)x0;
  int y1 = (y0 < NF - 1) ? y0 + 1 : NF - 1;
  int x1 = (x0 < NF - 1) ? x0 + 1 : NF - 1;
  const float* r0 = p + y0 * NF;
  const float* r1 = p + y1 * NF;
  float a = r0[x0] + fx * (r0[x1] - r0[x0]);
  float b = r1[x0] + fx * (r1[x1] - r1[x0]);
  return a + fy * (b - a);
}

__device__ __forceinline__ float sample_tri(const float* __restrict__ vol,
                                            float b1, float b2, float b3,
                                            int o, int y, int x, float ct, float st) {
  float dx = ct * b1 - st * b2;
  float dy = st * b1 + ct * b2;
  float co = fmodf((float)o - b3, 8.0f);
  co = (co < 0.0f) ? co + 8.0f : co;
  co = (co >= 8.0f) ? co - 8.0f : co;
  int o0 = (int)co;
  float fo = co - (float)o0;
  int o1 = (o0 + 1) & 7;                            // wrap-pad: xp[8] == x[0]
  float cy = fminf(fmaxf((float)y - dy, 0.0f), (float)(NF - 1));
  float cx = fminf(fmaxf((float)x - dx, 0.0f), (float)(NF - 1));
  float s0 = bilin(vol + (long long)o0 * HW2, cy, cx);
  float s1 = bilin(vol + (long long)o1 * HW2, cy, cx);
  return s0 + fo * (s1 - s0);
}

// ---------------------------------------------------------------------------
// Tables: DFT cos/sin matrices (F symmetric) + cakewavelet filters psi
// ---------------------------------------------------------------------------
__global__ void init_tables(float* __restrict__ Ct, float* __restrict__ St, float* __restrict__ psi) {
  long long idx = (long long)blockIdx.x * 256 + threadIdx.x;  // ORN*HW2 threads
  int ij = (int)(idx % HW2);
  int o  = (int)(idx / HW2);
  int i = ij / NF, j = ij % NF;
  if (o == 0) {
    int m = (i * j) % NF;
    double a = (2.0 * 3.14159265358979323846 / 192.0) * (double)m;
    Ct[ij] = (float)cos(a);
    St[ij] = (float)sin(a);
  }
  float fy = ((i < NF / 2) ? (float)i : (float)(i - NF)) / (float)NF;
  float fx = ((j < NF / 2) ? (float)j : (float)(j - NF)) / (float)NF;
  float rho = sqrtf(fx * fx + fy * fy);
  float phi = atan2f(fy, fx);
  float rr = rho / 0.4f;                            // rho / (0.5*inflection)
  float r2 = rr * rr, r4 = r2 * r2, r8 = r4 * r4;
  float radial = expf(-r8);
  float th = (float)o * DTH;
  float d = fmodf(phi - th + PI_F, TWO_PI_F);
  if (d < 0.0f) d += TWO_PI_F;
  d -= PI_F;
  float ang = fmaxf(0.0f, 1.0f - fabsf(d) / DTH);
  float v = ang * radial;
  if (i == 0 && j == 0) v = 0.125f;                 // split DC evenly over 8 orientations
  psi[(long long)o * HW2 + ij] = v;
}

__global__ void scale_psi(const float* __restrict__ psi, const float* __restrict__ xhre,
                          const float* __restrict__ xhim, float* __restrict__ yre,
                          float* __restrict__ yim) {
  long long idx = (long long)blockIdx.x * 256 + threadIdx.x;  // 16*HW2
  int ij = (int)(idx % HW2);
  int bo = (int)(idx / HW2);
  int b = bo >> 3, o = bo & 7;
  float p = psi[(long long)o * HW2 + ij];
  yre[idx] = p * xhre[(long long)b * HW2 + ij];
  yim[idx] = p * xhim[(long long)b * HW2 + ij];
}

__global__ void zero_d(double* p, int n) {
  int i = blockIdx.x * 256 + threadIdx.x;
  if (i < n) p[i] = 0.0;
}

// ---------------------------------------------------------------------------
// Lift head: BN1 statistics over the 40 per-channel shifted samples of the
// single-channel lift (lin1/convect/bn/lin2 folded into a 16x40 affine map).
// ---------------------------------------------------------------------------
__global__ __launch_bounds__(256) void lift_stats(const float* __restrict__ u0,
                                                  const float* __restrict__ gl,
                                                  double* __restrict__ stats) {
  int c = blockIdx.y;
  long long idx = (long long)blockIdx.x * 256 + threadIdx.x;  // PTSN
  int xc = (int)(idx % NF);
  long long t = idx / NF;
  int y = (int)(t % NF); t /= NF;
  int o = (int)(t % ORN);
  int b = (int)(t / ORN);
  float th = (float)o * DTH;
  float ct = cosf(th), st = sinf(th);
  float S = sample_tri(u0 + (long long)b * VOLN, gl[3 * c], gl[3 * c + 1], gl[3 * c + 2],
                       o, y, xc, ct, st);
  __shared__ double s1[256], s2[256];
  s1[threadIdx.x] = (double)S;
  s2[threadIdx.x] = (double)S * (double)S;
  __syncthreads();
  for (int w = 128; w > 0; w >>= 1) {
    if ((int)threadIdx.x < w) { s1[threadIdx.x] += s1[threadIdx.x + w]; s2[threadIdx.x] += s2[threadIdx.x + w]; }
    __syncthreads();
  }
  if (threadIdx.x == 0) { atomicAdd(&stats[c], s1[0]); atomicAdd(&stats[C1N + c], s2[0]); }
}

__global__ void bn1_prep(const double* __restrict__ stats,
                         const float* __restrict__ w1, const float* __restrict__ b1v,
                         const float* __restrict__ gamma, const float* __restrict__ beta,
                         const float* __restrict__ l2w, const float* __restrict__ l2b,
                         float* __restrict__ M2, float* __restrict__ B2) {
  __shared__ float effw[C1N], effb[C1N];
  int t = threadIdx.x;
  if (t < C1N) {
    double n = (double)PTSN;
    float mS = (float)(stats[t] / n);
    float vS = (float)(stats[C1N + t] / n) - mS * mS;
    float mean = w1[t] * mS + b1v[t];               // lin1 is Cin=1: v = w1*S + b1
    float var  = w1[t] * w1[t] * vS;
    float a = gamma[t] * rsqrtf(var + 1e-5f);
    effw[t] = a * w1[t];
    effb[t] = a * (b1v[t] - mean) + beta[t];
  }
  __syncthreads();
  if (t < C2N) {
    float bb = l2b[t];
    for (int c = 0; c < C1N; ++c) {
      M2[t * C1N + c] = l2w[t * C1N + c] * effw[c];
      bb += l2w[t * C1N + c] * effb[c];
    }
    B2[t] = bb;
  }
}

__global__ __launch_bounds__(256) void lift_apply(const float* __restrict__ u0,
                                                  const float* __restrict__ gl,
                                                  const float* __restrict__ M2,
                                                  const float* __restrict__ B2,
                                                  float* __restrict__ out16) {
  __shared__ float sM[C2N * C1N];
  __shared__ float sB[C2N];
  __shared__ float sG[C1N * 3];
  for (int i = threadIdx.x; i < C2N * C1N; i += 256) sM[i] = M2[i];
  if (threadIdx.x < C2N) sB[threadIdx.x] = B2[threadIdx.x];
  for (int i = threadIdx.x; i < C1N * 3; i += 256) sG[i] = gl[i];
  __syncthreads();
  long long idx = (long long)blockIdx.x * 256 + threadIdx.x;  // PTSN
  int xc = (int)(idx % NF);
  long long t = idx / NF;
  int y = (int)(t % NF); t /= NF;
  int o = (int)(t % ORN);
  int b = (int)(t / ORN);
  float th = (float)o * DTH;
  float ct = cosf(th), st = sinf(th);
  const float* vol = u0 + (long long)b * VOLN;
  float acc[C2N];
  #pragma unroll
  for (int d = 0; d < C2N; ++d) acc[d] = 0.0f;
  for (int c = 0; c < C1N; ++c) {
    float S = sample_tri(vol, sG[3 * c], sG[3 * c + 1], sG[3 * c + 2], o, y, xc, ct, st);
    #pragma unroll
    for (int d = 0; d < C2N; ++d) acc[d] += sM[d * C1N + c] * S;
  }
  long long base = ((long long)b * C2N) * VOLN + (long long)o * HW2 + (long long)y * NF + xc;
  #pragma unroll
  for (int d = 0; d < C2N; ++d) out16[base + (long long)d * VOLN] = acc[d] + sB[d];
}

// ---------------------------------------------------------------------------
// CDE building blocks
// ---------------------------------------------------------------------------
__global__ __launch_bounds__(256) void convect16(const float* __restrict__ in,
                                                 float* __restrict__ out,
                                                 const float* __restrict__ g) {
  long long idx = (long long)blockIdx.x * 256 + threadIdx.x;  // CDE_ELEMS
  int xc = (int)(idx % NF);
  long long t = idx / NF;
  int y = (int)(t % NF); t /= NF;
  int o = (int)(t % ORN); t /= ORN;
  int c = (int)(t % C2N);
  int b = (int)(t / C2N);
  float th = (float)o * DTH;
  float ct = cosf(th), st = sinf(th);
  out[idx] = sample_tri(in + (long long)(b * C2N + c) * VOLN,
                        g[c * 3 + 0], g[c * 3 + 1], g[c * 3 + 2], o, y, xc, ct, st);
}

__global__ void morph_prep(const float* __restrict__ w, float* __restrict__ ktab) {
  int i = blockIdx.x * 256 + threadIdx.x;
  if (i >= C2N * 125) return;
  int c = i / 125, off = i % 125;
  int b3 = off / 25 - 2, b1 = (off / 5) % 5 - 2, b2 = off % 5 - 2;
  float w0 = w[c * 3 + 0], w1 = w[c * 3 + 1], w2 = w[c * 3 + 2];
  float p0 = (w0 > 20.0f) ? w0 : log1pf(expf(w0));  // softplus
  float p1 = (w1 > 20.0f) ? w1 : log1pf(expf(w1));
  float p2 = (w2 > 20.0f) ? w2 : log1pf(expf(w2));
  float bd = (float)b3 * DTH;
  float dist2 = p0 * (float)(b1 * b1) + p1 * (float)(b2 * b2) + p2 * bd * bd;
  ktab[i] = powf(dist2, 2.1666666667f);             // nu/2, nu = 2a/(2a-1), a=0.65
}

template <bool DIL>
__global__ __launch_bounds__(256) void morph16(const float* __restrict__ in,
                                               float* __restrict__ out,
                                               const float* __restrict__ ktab) {
  __shared__ float kk[125];
  long long idx = (long long)blockIdx.x * 256 + threadIdx.x;  // CDE_ELEMS
  int xc = (int)(idx % NF);
  long long t = idx / NF;
  int y = (int)(t % NF); t /= NF;
  int o = (int)(t % ORN); t /= ORN;
  int c = (int)(t % C2N);
  int b = (int)(t / C2N);
  if (threadIdx.x < 125) kk[threadIdx.x] = ktab[c * 125 + threadIdx.x];  // c uniform per block
  __syncthreads();
  const float* vol = in + (long long)(b * C2N + c) * VOLN;
  float th = (float)o * DTH;
  float ct = cosf(th), st = sinf(th);
  float best = DIL ? -INFINITY : INFINITY;
  for (int off = 0; off < 125; ++off) {
    int b3 = off / 25 - 2;
    int b1 = (off / 5) % 5 - 2;
    int b2 = off % 5 - 2;
    int io = (o - b3 + 8) & 7;                      // integer b3 -> exact orientation index
    float dx = ct * (float)b1 - st * (float)b2;
    float dy = st * (float)b1 + ct * (float)b2;
    float cy = fminf(fmaxf((float)y - dy, 0.0f), (float)(NF - 1));
    float cx = fminf(fmaxf((float)xc - dx, 0.0f), (float)(NF - 1));
    float s = bilin(vol + (long long)io * HW2, cy, cx);
    float v = DIL ? (s - kk[off]) : (s + kk[off]);
    best = DIL ? fmaxf(best, v) : fminf(best, v);
  }
  out[idx] = best;
}

__global__ __launch_bounds__(256) void bn_stats16(const float* __restrict__ u, double* __restrict__ st) {
  int c = blockIdx.y;
  long long idx = (long long)blockIdx.x * 256 + threadIdx.x;  // PTSN points per channel
  int b = (int)(idx / VOLN);
  long long r = idx % VOLN;
  float v = u[(long long)(b * C2N + c) * VOLN + r];
  __shared__ double s1[256], s2[256];
  s1[threadIdx.x] = (double)v;
  s2[threadIdx.x] = (double)v * (double)v;
  __syncthreads();
  for (int w = 128; w > 0; w >>= 1) {
    if ((int)threadIdx.x < w) { s1[threadIdx.x] += s1[threadIdx.x + w]; s2[threadIdx.x] += s2[threadIdx.x + w]; }
    __syncthreads();
  }
  if (threadIdx.x == 0) { atomicAdd(&st[c], s1[0]); atomicAdd(&st[C2N + c], s2[0]); }
}

__global__ __launch_bounds__(256) void bn_apply16(float* __restrict__ u, const double* __restrict__ st,
                                                  const float* __restrict__ gamma,
                                                  const float* __restrict__ beta) {
  long long idx = (long long)blockIdx.x * 256 + threadIdx.x;  // CDE_ELEMS
  int c = (int)((idx / VOLN) % C2N);
  double n = (double)PTSN;
  float mean = (float)(st[c] / n);
  float var  = (float)(st[C2N + c] / n) - mean * mean;
  float a = gamma[c] * rsqrtf(var + 1e-5f);
  u[idx] = a * (u[idx] - mean) + beta[c];
}

__global__ __launch_bounds__(256) void final_head(const float* __restrict__ u,
                                                  const float* __restrict__ fw,
                                                  float* __restrict__ out) {
  int idx = blockIdx.x * 256 + threadIdx.x;         // NB*HW2
  int b = idx / HW2, r = idx % HW2;
  float dot = 0.0f;
  for (int d = 0; d < C2N; ++d) {
    const float* p = u + (long long)(b * C2N + d) * VOLN + r;
    float m = p[0];
    #pragma unroll
    for (int o = 1; o < ORN; ++o) m = fmaxf(m, p[(long long)o * HW2]);
    dot += fw[d] * m;
  }
  out[idx] = 1.0f / (1.0f + expf(-dot));
}

// ---------------------------------------------------------------------------
static inline void launch_gemm(hipStream_t stream, const float* A1, const float* A2,
                               const float* B1, const float* B2, const float* bias, float* D,
                               int M, int N, int K, int lda, int ldb, int ldd,
                               long long sA, long long sB, long long sD,
                               float s2, float scale, int batch) {
  GemmArgs g{A1, A2, B1, B2, bias, D, K, lda, ldb, ldd, sA, sB, sD, s2, scale};
  int tm = M / 16, tn = N / 16;
  int waves = tm * tn * batch;
  int blocks = (waves + 3) / 4;
  gemm2_wmma<<<blocks, 128, 0, stream>>>(g, tm, tn, batch);
}

extern "C" void kernel_launch(void* const* d_in, const int* in_sizes, int n_in,
                              void* d_out, int out_size, void* d_ws, size_t ws_size,
                              hipStream_t stream) {
  (void)in_sizes; (void)n_in; (void)out_size;
  if (ws_size < (size_t)WS_FLOATS * sizeof(float)) return;   // deterministic no-op guard

  const float* x      = (const float*)d_in[0];   // [2,1,192,192]
  const float* lin1_w = (const float*)d_in[1];   // [40,1]
  const float* lin1_b = (const float*)d_in[2];   // [40]
  const float* g_lift = (const float*)d_in[3];   // [40,3]
  const float* bn1_g  = (const float*)d_in[4];   // [40]
  const float* bn1_b  = (const float*)d_in[5];   // [40]
  const float* lin2_w = (const float*)d_in[6];   // [16,40]
  const float* lin2_b = (const float*)d_in[7];   // [16]
  const float* cde_lw = (const float*)d_in[8];   // [4,16,16]
  const float* cde_lb = (const float*)d_in[9];   // [4,16]
  const float* cde_g  = (const float*)d_in[10];  // [4,16,3]
  const float* cde_wd = (const float*)d_in[11];  // [4,16,3]
  const float* cde_we = (const float*)d_in[12];  // [4,16,3]
  const float* cde_bg = (const float*)d_in[13];  // [4,16]
  const float* cde_bb = (const float*)d_in[14];  // [4,16]
  const float* fin_w  = (const float*)d_in[15];  // [1,16]
  float* out = (float*)d_out;

  float* ws   = (float*)d_ws;
  float* Ct   = ws + OFF_C;
  float* St   = ws + OFF_S;
  float* psi  = ws + OFF_PSI;
  float* u0   = ws + OFF_U0;
  float* Tre  = ws + OFF_TRE;
  float* Tim  = ws + OFF_TIM;
  float* XHre = ws + OFF_XHRE;
  float* XHim = ws + OFF_XHIM;
  float* Yre  = ws + OFF_YRE;
  float* Yim  = ws + OFF_YIM;
  float* Pb   = ws + OFF_P;
  float* Qb   = ws + OFF_Q;
  double* stats = (double*)(ws + OFF_STAT);      // [0..79]: lift sums; +128: CDE sums
  double* st16  = stats + 128;
  float* M2   = ws + OFF_M2;
  float* B2v  = ws + OFF_B2;
  float* ktab = ws + OFF_KTAB;
  float* bufA = ws + OFF_BUFA;
  float* bufB = ws + OFF_BUFB;

  // ---- tables (DFT matrices + cake filters) ----
  init_tables<<<(ORN * HW2) / 256, 256, 0, stream>>>(Ct, St, psi);

  // ---- fft2(x) as GEMMs: Xh = F X F, F = C - iS (symmetric) ----
  launch_gemm(stream, x, nullptr, Ct, nullptr, nullptr, Tre, NF, NF, NF, NF, NF, NF,
              HW2, 0, HW2, 1.f, 1.f, NB);                        // T_re = X C
  launch_gemm(stream, x, nullptr, St, nullptr, nullptr, Tim, NF, NF, NF, NF, NF, NF,
              HW2, 0, HW2, 1.f, -1.f, NB);                       // T_im = -(X S)
  launch_gemm(stream, Ct, St, Tre, Tim, nullptr, XHre, NF, NF, NF, NF, NF, NF,
              0, HW2, HW2, 1.f, 1.f, NB);                        // Xh_re = C Tre + S Tim
  launch_gemm(stream, Ct, St, Tim, Tre, nullptr, XHim, NF, NF, NF, NF, NF, NF,
              0, HW2, HW2, -1.f, 1.f, NB);                       // Xh_im = C Tim - S Tre

  // ---- per-orientation filter + ifft2 (real part) ----
  scale_psi<<<(16 * HW2) / 256, 256, 0, stream>>>(psi, XHre, XHim, Yre, Yim);
  launch_gemm(stream, Ct, St, Yre, Yim, nullptr, Pb, NF, NF, NF, NF, NF, NF,
              0, HW2, HW2, -1.f, 1.f, 16);                       // P = C Yre - S Yim
  launch_gemm(stream, Ct, St, Yim, Yre, nullptr, Qb, NF, NF, NF, NF, NF, NF,
              0, HW2, HW2, 1.f, 1.f, 16);                        // Q = C Yim + S Yre
  launch_gemm(stream, Pb, Qb, Ct, St, nullptr, u0, NF, NF, NF, NF, NF, NF,
              HW2, 0, HW2, -1.f, 1.0f / (float)HW2, 16);         // u0 = (P C - Q S)/N^2

  // ---- lift head: lin1 -> convect -> bn -> lin2 (folded) ----
  zero_d<<<1, 256, 0, stream>>>(stats, 256);
  lift_stats<<<dim3(PTSN / 256, C1N), 256, 0, stream>>>(u0, g_lift, stats);
  bn1_prep<<<1, 64, 0, stream>>>(stats, lin1_w, lin1_b, bn1_g, bn1_b, lin2_w, lin2_b, M2, B2v);
  lift_apply<<<PTSN / 256, 256, 0, stream>>>(u0, g_lift, M2, B2v, bufA);

  // ---- 4 CDE layers: linear (WMMA) -> convect -> dilate -> erode -> BN ----
  const long long CH_BATCH = (long long)C2N * VOLN;
  for (int i = 0; i < 4; ++i) {
    launch_gemm(stream, cde_lw + i * C2N * C2N, nullptr, bufA, nullptr,
                cde_lb + i * C2N, bufB,
                C2N, VOLN, C2N, C2N, VOLN, VOLN, 0, CH_BATCH, CH_BATCH, 1.f, 1.f, NB);
    convect16<<<(int)(CDE_ELEMS / 256), 256, 0, stream>>>(bufB, bufA, cde_g + i * C2N * 3);
    morph_prep<<<8, 256, 0, stream>>>(cde_wd + i * C2N * 3, ktab);
    morph16<true><<<(int)(CDE_ELEMS / 256), 256, 0, stream>>>(bufA, bufB, ktab);
    morph_prep<<<8, 256, 0, stream>>>(cde_we + i * C2N * 3, ktab);
    morph16<false><<<(int)(CDE_ELEMS / 256), 256, 0, stream>>>(bufB, bufA, ktab);
    zero_d<<<1, 256, 0, stream>>>(st16, 32);
    bn_stats16<<<dim3(PTSN / 256, C2N), 256, 0, stream>>>(bufA, st16);
    bn_apply16<<<(int)(CDE_ELEMS / 256), 256, 0, stream>>>(bufA, st16,
                                                           cde_bg + i * C2N, cde_bb + i * C2N);
  }

  // ---- max over orientations + 1x1 conv + sigmoid ----
  final_head<<<(NB * HW2) / 256, 256, 0, stream>>>(bufA, fin_w, out);
}